// BCOP_52407190946192
// MI455X (gfx1250) — compile-verified
//
#include <hip/hip_runtime.h>

#define MSZ 65536  // 256*256 elements per matrix

typedef __attribute__((ext_vector_type(2))) float v2f;
typedef __attribute__((ext_vector_type(8))) float v8f;

__device__ __forceinline__ v8f wmma4(v2f a, v2f b, v8f c) {
  // D = A(16x4) * B(4x16) + C(16x16), fp32
  return __builtin_amdgcn_wmma_f32_16x16x4_f32(false, a, false, b, (short)0, c,
                                               false, false);
}

// ---------------- wave-level 16x16-tile GEMM helpers (256x256 row-major) ----
// A-frag layout (16x4 f32): lane m=L&15 is row M; half=L>>4 selects K pair:
//   a.x = A[m, 2*half], a.y = A[m, 2*half+1]
// B-frag layout (4x16 f32): b.x = B[2*half, n], b.y = B[2*half+1, n], n=L&15
// C/D (16x16 f32): elem r of v8f -> row = 8*half + r, col = L&15

template <typename BP>
__device__ __forceinline__ v8f wave_gemm_nn(const float* A, BP B, int tm,
                                            int tn, int K, v8f acc) {
  const int lane = threadIdx.x & 31;
  const int mn = lane & 15;
  const int half = lane >> 4;
  const float* ap = A + (tm * 16 + mn) * 256 + half * 2;
  BP bp = B + (half * 2) * 256 + tn * 16 + mn;
  for (int k = 0; k < K; k += 4) {
    v2f a = *(const v2f*)ap;          // contiguous b64
    v2f b = {bp[0], bp[256]};         // two strided b32
    acc = wmma4(a, b, acc);
    ap += 4;
    bp += 1024;
  }
  return acc;
}

__device__ __forceinline__ v8f wave_gemm_tn(const float* A, const float* B,
                                            int tm, int tn, int K, v8f acc) {
  // C = A^T * B
  const int lane = threadIdx.x & 31;
  const int mn = lane & 15;
  const int half = lane >> 4;
  const float* ap = A + (half * 2) * 256 + tm * 16 + mn;
  const float* bp = B + (half * 2) * 256 + tn * 16 + mn;
  for (int k = 0; k < K; k += 4) {
    v2f a = {ap[0], ap[256]};
    v2f b = {bp[0], bp[256]};
    acc = wmma4(a, b, acc);
    ap += 1024;
    bp += 1024;
  }
  return acc;
}

__device__ __forceinline__ v8f wave_gemm_nt(const float* A, const float* B,
                                            int tm, int tn, int K, v8f acc) {
  // C = A * B^T
  const int lane = threadIdx.x & 31;
  const int mn = lane & 15;
  const int half = lane >> 4;
  const float* ap = A + (tm * 16 + mn) * 256 + half * 2;
  const float* bp = B + (tn * 16 + mn) * 256 + half * 2;
  for (int k = 0; k < K; k += 4) {
    v2f a = *(const v2f*)ap;
    v2f b = *(const v2f*)bp;
    acc = wmma4(a, b, acc);
    ap += 4;
    bp += 4;
  }
  return acc;
}

template <typename CP>
__device__ __forceinline__ void store_tile(CP C, int tm, int tn, v8f acc) {
  const int lane = threadIdx.x & 31;
  const int n = lane & 15;
  const int half = lane >> 4;
  CP cp = C + (tm * 16 + half * 8) * 256 + tn * 16 + n;
#pragma unroll
  for (int r = 0; r < 8; ++r) cp[r * 256] = acc[r];
}

// ---------------- 1) power iteration + scale: wa[b] = pm[b] / sigma_max -----
__global__ __launch_bounds__(256) void pi_kernel(const float* __restrict__ pm,
                                                 float* __restrict__ wa) {
  const int b = blockIdx.x;
  const float* A = pm + (size_t)b * MSZ;
  __shared__ float u[256], v[256], red[256];
  const int t = threadIdx.x;
  u[t] = sinf(0.37f * (float)t + 1.3f * (float)b) + 0.01f;  // deterministic init
  __syncthreads();
  for (int it = 0; it < 10; ++it) {
    float s = 0.0f;                       // v = A^T u
    for (int i = 0; i < 256; ++i) s += A[i * 256 + t] * u[i];
    red[t] = s * s;
    __syncthreads();
    for (int off = 128; off > 0; off >>= 1) {
      if (t < off) red[t] += red[t + off];
      __syncthreads();
    }
    const float nv = sqrtf(red[0]);
    __syncthreads();
    v[t] = s / nv;
    __syncthreads();
    float s2 = 0.0f;                      // u = A v
    for (int j = 0; j < 256; ++j) s2 += A[t * 256 + j] * v[j];
    red[t] = s2 * s2;
    __syncthreads();
    for (int off = 128; off > 0; off >>= 1) {
      if (t < off) red[t] += red[t + off];
      __syncthreads();
    }
    const float nu = sqrtf(red[0]);
    __syncthreads();
    u[t] = s2 / nu;
    __syncthreads();
  }
  float ti = 0.0f;                        // sigma = u^T A v
  for (int j = 0; j < 256; ++j) ti += A[t * 256 + j] * v[j];
  red[t] = u[t] * ti;
  __syncthreads();
  for (int off = 128; off > 0; off >>= 1) {
    if (t < off) red[t] += red[t + off];
    __syncthreads();
  }
  const float inv = 1.0f / red[0];
  float* wp = wa + (size_t)b * MSZ;
  for (int i = t; i < MSZ; i += 256) wp[i] = A[i] * inv;
}

// ------- 2) fused 20 Bjorck iterations; wtw lives in 256KB LDS -------------
__global__ __launch_bounds__(1024) void bjorck_kernel(float* wa, float* wb) {
  extern __shared__ float wtw[];          // 65536 floats = 256 KB (<=320KB WGP LDS)
  const int bt = blockIdx.x;
  float* p = wa + (size_t)bt * MSZ;
  float* q = wb + (size_t)bt * MSZ;
  const int wv = threadIdx.x >> 5;        // 0..31 waves
  const int lane = threadIdx.x & 31;
  const int n = lane & 15;
  const int half = lane >> 4;

  for (int it = 0; it < 20; ++it) {
    // phase 1: wtw = p^T p  (each wave: 8 of the 256 tiles)
    for (int t8 = 0; t8 < 8; ++t8) {
      const int t = wv * 8 + t8;
      const int tm = t >> 4, tn = t & 15;
      v8f acc = {};
      acc = wave_gemm_tn(p, p, tm, tn, 256, acc);
      store_tile((float*)wtw, tm, tn, acc);
    }
    __syncthreads();
    // phase 2: q = 1.5*p - 0.5*(p @ wtw)
    for (int t8 = 0; t8 < 8; ++t8) {
      const int t = wv * 8 + t8;
      const int tm = t >> 4, tn = t & 15;
      v8f acc = {};
      acc = wave_gemm_nn(p, (const float*)wtw, tm, tn, 256, acc);
      float* qp = q + (tm * 16 + half * 8) * 256 + tn * 16 + n;
      const float* pp = p + (tm * 16 + half * 8) * 256 + tn * 16 + n;
#pragma unroll
      for (int r = 0; r < 8; ++r)
        qp[r * 256] = 1.5f * pp[r * 256] - 0.5f * acc[r];
    }
    __threadfence_block();
    __syncthreads();
    float* tmp = p; p = q; q = tmp;       // 20 (even) swaps -> result in wa
  }
}

// ------- 3) PQ[m] = (ortho[1+m][:, :128]) @ (ortho[1+m][:, :128])^T --------
__global__ void pq_kernel(const float* __restrict__ wa, float* __restrict__ pq) {
  const int mi = blockIdx.z;
  const float* A = wa + (size_t)(1 + mi) * MSZ;
  v8f acc = {};
  acc = wave_gemm_nt(A, A, blockIdx.x, blockIdx.y, 128, acc);  // mask => K=128
  store_tile(pq + (size_t)mi * MSZ, blockIdx.x, blockIdx.y, acc);
}

// ------- 4) generic C = A @ B (one wave per 16x16 tile) --------------------
__global__ void gemm_nn_kernel(const float* __restrict__ A,
                               const float* __restrict__ B,
                               float* __restrict__ C) {
  v8f acc = {};
  acc = wave_gemm_nn(A, B, blockIdx.x, blockIdx.y, 256, acc);
  store_tile(C, blockIdx.x, blockIdx.y, acc);
}

// ------- 5) elementwise: HF[a][b] and S[c][d] from block_orth algebra ------
// block_orth entries are affine combos of {I, P, P', P@P'} -> only 1 GEMM each
__global__ void combine_kernel(const float* __restrict__ H,
                               const float* __restrict__ HM1,
                               const float* __restrict__ HP0,
                               const float* __restrict__ HP1,
                               const float* __restrict__ M2,
                               const float* __restrict__ P2,
                               const float* __restrict__ P3,
                               float* __restrict__ HF, float* __restrict__ S) {
  const int idx = blockIdx.x * 256 + threadIdx.x;
  const int row = idx >> 8, col = idx & 255;
  const float hm1 = HM1[idx], hp0 = HP0[idx], hp1 = HP1[idx], h = H[idx];
  HF[idx]           = hm1;                    // H @ (P0 P1)
  HF[MSZ + idx]     = hp0 - hm1;              // H @ (P0 (I-P1))
  HF[2 * MSZ + idx] = hp1 - hm1;              // H @ ((I-P0) P1)
  HF[3 * MSZ + idx] = h - hp0 - hp1 + hm1;    // H @ ((I-P0)(I-P1))
  const float m2 = M2[idx], p2 = P2[idx], p3 = P3[idx];
  const float eye = (row == col) ? 1.0f : 0.0f;
  S[idx]           = m2;
  S[MSZ + idx]     = p2 - m2;
  S[2 * MSZ + idx] = p3 - m2;
  S[3 * MSZ + idx] = eye - p2 - p3 + m2;
}

// ------- 6) matrix_conv: T[k][l] = sum HF[i1][j1] @ S[k-i1][l-j1]; store
//            transposed per-tap weight Wt[k*3+l][out_ch j][in_ch i] ---------
__global__ void ttap_kernel(const float* __restrict__ HF,
                            const float* __restrict__ S,
                            float* __restrict__ Wt) {
  const int z = blockIdx.z;
  const int k = z / 3, l = z % 3;
  const int tm = blockIdx.x, tn = blockIdx.y;
  v8f acc = {};
  for (int i1 = 0; i1 < 2; ++i1) {
    const int i2 = k - i1;
    if (i2 < 0 || i2 > 1) continue;
    for (int j1 = 0; j1 < 2; ++j1) {
      const int j2 = l - j1;
      if (j2 < 0 || j2 > 1) continue;
      acc = wave_gemm_nn(HF + (size_t)(i1 * 2 + j1) * MSZ,
                         S + (size_t)(i2 * 2 + j2) * MSZ, tm, tn, 256, acc);
    }
  }
  const int lane = threadIdx.x & 31;
  const int n = lane & 15;
  const int half = lane >> 4;
  float* wp = Wt + (size_t)z * MSZ;
#pragma unroll
  for (int r = 0; r < 8; ++r) {
    const int i = tm * 16 + half * 8 + r;   // in-channel
    const int j = tn * 16 + n;              // out-channel
    wp[j * 256 + i] = acc[r];               // transposed store for conv A-side
  }
}

// ------- 7) circular 3x3 conv as 9-tap WMMA GEMM ---------------------------
// out[b,j,y,x] = bias[j] + sum_{l,k,i} x[b,i,(y+l-1)&63,(x+k-1)&63] * Wt[k*3+l][j][i]
// Each wave: 4 j-tiles x 4 x-tiles (64 out-ch x 64 x) -> 16 WMMAs per 12 loads.
// Block = 8 waves = 2 y-rows x 256 out-channels. Grid = (32 y-pairs, 16 batch).
__global__ __launch_bounds__(256) void conv_kernel(
    const float* __restrict__ x, const float* __restrict__ wt,
    const float* __restrict__ bias, float* __restrict__ out) {
  const int b = blockIdx.y;                    // 0..15
  const int wv = threadIdx.x >> 5;             // 0..7
  const int y = blockIdx.x * 2 + (wv >> 2);    // 0..63
  const int jbase = (wv & 3) * 64;             // 64 out-channels per wave
  const int lane = threadIdx.x & 31;
  const int mn = lane & 15;
  const int half = lane >> 4;

  v8f acc[4][4];
#pragma unroll
  for (int mt = 0; mt < 4; ++mt) {
    v8f binit;
#pragma unroll
    for (int r = 0; r < 8; ++r)
      binit[r] = bias[jbase + mt * 16 + half * 8 + r];
#pragma unroll
    for (int t = 0; t < 4; ++t) acc[mt][t] = binit;
  }

  const float* xb = x + (size_t)b * 1048576;   // x[b][ch][yy][xx]

  for (int l = 0; l < 3; ++l) {
    const int yl = (y + l + 63) & 63;
    const float* xrow = xb + yl * 64 + (half * 2) * 4096;
    for (int k = 0; k < 3; ++k) {
      const float* wtap = wt + (k * 3 + l) * MSZ;
      const float* ap[4];
#pragma unroll
      for (int mt = 0; mt < 4; ++mt)
        ap[mt] = wtap + (jbase + mt * 16 + mn) * 256 + half * 2;  // b64 loads
      const float* bp[4];
#pragma unroll
      for (int t = 0; t < 4; ++t)
        bp[t] = xrow + ((t * 16 + mn + k + 63) & 63);
      for (int i0 = 0; i0 < 64; ++i0) {        // K = 256 channels, step 4
        v2f a[4];
#pragma unroll
        for (int mt = 0; mt < 4; ++mt) {
          a[mt] = *(const v2f*)ap[mt];
          ap[mt] += 4;
        }
#pragma unroll
        for (int t = 0; t < 4; ++t) {
          v2f bb = {bp[t][0], bp[t][4096]};
          bp[t] += 16384;                      // advance 4 channels
#pragma unroll
          for (int mt = 0; mt < 4; ++mt)
            acc[mt][t] = wmma4(a[mt], bb, acc[mt][t]);
        }
      }
    }
  }

  float* op = out + (size_t)b * 1048576 + (size_t)y * 64;
#pragma unroll
  for (int mt = 0; mt < 4; ++mt)
#pragma unroll
    for (int t = 0; t < 4; ++t)
#pragma unroll
      for (int r = 0; r < 8; ++r) {
        const int j = jbase + mt * 16 + half * 8 + r;
        op[(size_t)j * 4096 + t * 16 + mn] = acc[mt][t][r];
      }
}

// ---------------------------------------------------------------------------
extern "C" void kernel_launch(void* const* d_in, const int* in_sizes, int n_in,
                              void* d_out, int out_size, void* d_ws,
                              size_t ws_size, hipStream_t stream) {
  (void)in_sizes; (void)n_in; (void)out_size; (void)ws_size;
  const float* x = (const float*)d_in[0];     // (16,256,64,64)
  const float* pm = (const float*)d_in[1];    // (5,256,256)
  const float* bias = (const float*)d_in[2];  // (256,)
  float* out = (float*)d_out;
  float* ws = (float*)d_ws;

  float* wa  = ws;                        // 5*MSZ  ortho result; [0] = H
  float* wb  = ws + 5 * (size_t)MSZ;      // 5*MSZ  Bjorck ping-pong
  float* PQ  = ws + 10 * (size_t)MSZ;     // 4*MSZ
  float* M1  = ws + 14 * (size_t)MSZ;
  float* M2  = ws + 15 * (size_t)MSZ;
  float* HM1 = ws + 16 * (size_t)MSZ;
  float* HP0 = ws + 17 * (size_t)MSZ;
  float* HP1 = ws + 18 * (size_t)MSZ;
  float* HF  = ws + 19 * (size_t)MSZ;     // 4*MSZ
  float* S   = ws + 23 * (size_t)MSZ;     // 4*MSZ
  float* Wt  = ws + 27 * (size_t)MSZ;     // 9*MSZ  (total 36*MSZ*4B = 9.4 MB)

  pi_kernel<<<5, 256, 0, stream>>>(pm, wa);
  bjorck_kernel<<<5, 1024, MSZ * sizeof(float), stream>>>(wa, wb);
  pq_kernel<<<dim3(16, 16, 4), 32, 0, stream>>>(wa, PQ);
  gemm_nn_kernel<<<dim3(16, 16), 32, 0, stream>>>(PQ, PQ + MSZ, M1);
  gemm_nn_kernel<<<dim3(16, 16), 32, 0, stream>>>(PQ + 2 * MSZ, PQ + 3 * MSZ, M2);
  gemm_nn_kernel<<<dim3(16, 16), 32, 0, stream>>>(wa, M1, HM1);
  gemm_nn_kernel<<<dim3(16, 16), 32, 0, stream>>>(wa, PQ, HP0);
  gemm_nn_kernel<<<dim3(16, 16), 32, 0, stream>>>(wa, PQ + MSZ, HP1);
  combine_kernel<<<256, 256, 0, stream>>>(wa, HM1, HP0, HP1, M2, PQ + 2 * MSZ,
                                          PQ + 3 * MSZ, HF, S);
  ttap_kernel<<<dim3(16, 16, 9), 32, 0, stream>>>(HF, S, Wt);
  conv_kernel<<<dim3(32, 16), 256, 0, stream>>>(x, Wt, bias, out);
}